// SelectiveSSMBlock_74526272520979
// MI455X (gfx1250) — compile-verified
//
#include <hip/hip_runtime.h>
#include <stdint.h>

// ---------------------------------------------------------------------------
// Selective SSM block for MI455X (gfx1250, wave32).
//   BATCH=4, SEQ=2048, D=768, N=16.
// Plan:
//   K1 proj_kernel  : Bp = u*W_B^T, Cp = u*W_C^T via V_WMMA_F32_16X16X4_F32.
//                     Weights staged in LDS (async-to-LDS), 4 row-tiles per
//                     wave for operand reuse. delta = softplus(p + u.q) fused.
//   K2 scan_pass1   : per chunk (128 steps): P[n] = prod exp(d*A), S = local scan.
//   K3 chunk_combine: sequential combine over 16 chunks -> per-chunk init X0.
//   K4 scan_pass2   : replay chunks with X0, y = x . C, write out (b,l,d).
// ---------------------------------------------------------------------------

#define BATCH   4
#define SEQ     2048
#define DMODEL  768
#define NSTATE  16
#define BL      (BATCH * SEQ)        // 8192 rows
#define CHUNK   128
#define NCHUNK  (SEQ / CHUNK)        // 16
#define DTILE   256
#define WSTRIDE (DMODEL + 2)         // LDS row padding: 770 % 64 != 0 -> no bank conflicts
#define TPW     4                    // row-tiles per wave in proj kernel

typedef __attribute__((ext_vector_type(2))) float v2f;
typedef __attribute__((ext_vector_type(8))) float v8f;

// ---- LDS helpers ----------------------------------------------------------
__device__ __forceinline__ unsigned lds_off(const void* p) {
  // addrspace(3) aperture: low 32 bits are the LDS byte offset
  return (unsigned)(uintptr_t)p;
}

// Async global->LDS staging (ASYNCcnt path). GV addressing: vaddr is 64-bit.
__device__ __forceinline__ void async_ld_b32(unsigned lds_byte, const void* g) {
  asm volatile("global_load_async_to_lds_b32 %0, %1, off"
               :: "v"(lds_byte), "v"((unsigned long long)(uintptr_t)g)
               : "memory");
}
__device__ __forceinline__ void async_ld_b128(unsigned lds_byte, const void* g) {
  asm volatile("global_load_async_to_lds_b128 %0, %1, off"
               :: "v"(lds_byte), "v"((unsigned long long)(uintptr_t)g)
               : "memory");
}
__device__ __forceinline__ void wait_async0() {
  asm volatile("s_wait_asynccnt 0" ::: "memory");
}

// ---------------------------------------------------------------------------
// K1: projections + delta.
// One wave computes FOUR 16x16 tiles of Bp and Cp (4x16 consecutive (b,l)
// rows, 16 states), marching K=768 in steps of 4 with V_WMMA_F32_16X16X4_F32.
// Weight B-fragments come from LDS and are reused across the 4 tiles:
// per K-step: 2 ds_load_b64 + 4 global_load_b64 -> 8 WMMAs (4 independent
// accumulation chains => software-pipelinable).
// A-frag (16x4 f32): lane m=lane&15 holds row m; khalf=lane>>4 selects K pair.
// B-frag (4x16 f32): lane holds col n=lane&15, rows 2*khalf, 2*khalf+1.
// ---------------------------------------------------------------------------
__global__ __launch_bounds__(128)
void proj_kernel(const float* __restrict__ u,      // (BL, D)
                 const float* __restrict__ W_B,    // (N, D)
                 const float* __restrict__ W_C,    // (N, D)
                 const float* __restrict__ q_delta,// (D)
                 const float* __restrict__ p_delta,// (1)
                 float* __restrict__ Bp,           // (BL, N)
                 float* __restrict__ Cp,           // (BL, N)
                 float* __restrict__ delta)        // (BL)
{
  __shared__ float sWB[NSTATE * WSTRIDE];   // ~49.3 KB
  __shared__ float sWC[NSTATE * WSTRIDE];   // ~49.3 KB
  __shared__ float sQ[DMODEL];              //   3 KB   (total ~102 KB < 320 KB)

  const int tid = threadIdx.x;

  // stage W_B / W_C (row-wise, honoring LDS padding) + q via ASYNCcnt path
  for (int i = tid; i < NSTATE * (DMODEL / 4); i += 128) {
    const int n = i / (DMODEL / 4);
    const int j = (i % (DMODEL / 4)) * 4;
    async_ld_b128(lds_off(&sWB[n * WSTRIDE + j]), W_B + (size_t)n * DMODEL + j);
    async_ld_b128(lds_off(&sWC[n * WSTRIDE + j]), W_C + (size_t)n * DMODEL + j);
  }
  for (int i = tid; i < DMODEL / 4; i += 128)
    async_ld_b128(lds_off(&sQ[i * 4]), q_delta + i * 4);
  wait_async0();
  __syncthreads();

  const int lane  = tid & 31;
  const int wave  = tid >> 5;
  const int tile0 = (blockIdx.x * 4 + wave) * TPW;  // 512 tiles total, 4/wave
  const int m     = lane & 15;                      // A row / B col
  const int khalf = lane >> 4;                      // 0 or 1

  const float* ur[TPW];
#pragma unroll
  for (int t = 0; t < TPW; ++t)
    ur[t] = u + (size_t)((tile0 + t) * 16 + m) * DMODEL;

  const float* wb = &sWB[m * WSTRIDE];
  const float* wc = &sWC[m * WSTRIDE];

  v8f   accB[TPW] = {};
  v8f   accC[TPW] = {};
  float dacc[TPW] = {};

#if __has_builtin(__builtin_amdgcn_wmma_f32_16x16x4_f32)
  for (int k0 = 0; k0 < DMODEL; k0 += 4) {
    const int ka = k0 + 2 * khalf;
    v2f bB = { wb[ka], wb[ka + 1] };       // ds_load_b64, shared by 4 tiles
    v2f bC = { wc[ka], wc[ka + 1] };
    const float q0 = sQ[ka], q1 = sQ[ka + 1];
#pragma unroll
    for (int t = 0; t < TPW; ++t) {
      v2f a = { ur[t][ka], ur[t][ka + 1] };
      accB[t] = __builtin_amdgcn_wmma_f32_16x16x4_f32(false, a, false, bB,
                                                      0, accB[t], false, false);
      accC[t] = __builtin_amdgcn_wmma_f32_16x16x4_f32(false, a, false, bC,
                                                      0, accC[t], false, false);
      dacc[t] += a.x * q0 + a.y * q1;
    }
  }
#else
  // Fallback (compile-safe scalar path; primary path is the WMMA one)
  for (int k0 = 0; k0 < DMODEL; k0 += 4) {
    const int ka = k0 + 2 * khalf;
    const float q0 = sQ[ka], q1 = sQ[ka + 1];
#pragma unroll
    for (int t = 0; t < TPW; ++t) {
      float ax = ur[t][ka], ay = ur[t][ka + 1];
      accB[t][0] = __fmaf_rn(ax, wb[ka], __fmaf_rn(ay, wb[ka + 1], accB[t][0]));
      accC[t][0] = __fmaf_rn(ax, wc[ka], __fmaf_rn(ay, wc[ka + 1], accC[t][0]));
      dacc[t] += ax * q0 + ay * q1;
    }
  }
#endif

  // C/D layout: VGPR i -> row (i + 8*khalf), col = lane&15
  const int nOut = lane & 15;
#pragma unroll
  for (int t = 0; t < TPW; ++t) {
    const int mBase = (tile0 + t) * 16 + (khalf ? 8 : 0);
#pragma unroll
    for (int i = 0; i < 8; ++i) {
      Bp[(size_t)(mBase + i) * NSTATE + nOut] = accB[t][i];
      Cp[(size_t)(mBase + i) * NSTATE + nOut] = accC[t][i];
    }
    // delta: lanes m and m+16 hold complementary K-partials of row m
    const float dsum = dacc[t] + __shfl_xor(dacc[t], 16, 32);
    if (lane < 16) {
      const float x  = p_delta[0] + dsum;
      const float sp = (x > 20.0f) ? x : log1pf(__expf(x));
      delta[(tile0 + t) * 16 + lane] = sp;
    }
  }
}

// ---------------------------------------------------------------------------
// K2: per-chunk local scan.  Block = (d-tile of 256, chunk c, batch b).
// Each thread owns one d channel and its 16-state recurrence over 128 steps.
// delta / Bp for the chunk are staged into LDS with async-to-LDS loads.
// ---------------------------------------------------------------------------
__global__ __launch_bounds__(256)
void scan_pass1(const float* __restrict__ u,     // (B, L, D)
                const float* __restrict__ A,     // (D, N)
                const float* __restrict__ delta, // (BL)
                const float* __restrict__ Bp,    // (BL, N)
                float* __restrict__ Pout,        // (B, NC, D, N)
                float* __restrict__ Sout)        // (B, NC, D, N)
{
  __shared__ float sDelta[CHUNK];
  __shared__ float sB[CHUNK * NSTATE];

  const int tid = threadIdx.x;
  const int dt  = blockIdx.x;            // 0..2
  const int c   = blockIdx.y;            // 0..15
  const int b   = blockIdx.z;            // 0..3
  const int d   = dt * DTILE + tid;
  const int l0  = c * CHUNK;
  const size_t rbase = (size_t)b * SEQ + l0;

  // stage chunk-wide delta (128 f32) and Bp (128x16 f32) via ASYNCcnt path
  if (tid < CHUNK)
    async_ld_b32(lds_off(&sDelta[0]) + tid * 4u, delta + rbase + tid);
  {
    const float* gB = Bp + rbase * NSTATE;
    for (int i = tid; i < (CHUNK * NSTATE) / 4; i += DTILE)
      async_ld_b128(lds_off(&sB[0]) + i * 16u, gB + i * 4);
  }
  wait_async0();
  __syncthreads();

  float Arow[NSTATE];
  {
    const float* Ad = A + (size_t)d * NSTATE;
#pragma unroll
    for (int n = 0; n < NSTATE; ++n) Arow[n] = Ad[n];
  }

  float P[NSTATE], S[NSTATE];
#pragma unroll
  for (int n = 0; n < NSTATE; ++n) { P[n] = 1.0f; S[n] = 0.0f; }

  const float* uptr = u + rbase * DMODEL + d;
  for (int t = 0; t < CHUNK; ++t) {
    __builtin_prefetch(uptr + (size_t)(t + 16) * DMODEL, 0, 0);
    const float dl = sDelta[t];
    const float du = dl * uptr[(size_t)t * DMODEL];
    const float* Bt = &sB[t * NSTATE];
#pragma unroll
    for (int n = 0; n < NSTATE; ++n) {
      float dA = __expf(dl * Arow[n]);           // v_exp_f32 (TRANS, co-exec)
      P[n] *= dA;
      S[n]  = __fmaf_rn(dA, S[n], du * Bt[n]);
    }
  }

  const size_t obase = (((size_t)b * NCHUNK + c) * DMODEL + d) * NSTATE;
#pragma unroll
  for (int n = 0; n < NSTATE; ++n) {
    Pout[obase + n] = P[n];
    Sout[obase + n] = S[n];
  }
}

// ---------------------------------------------------------------------------
// K3: sequential combine across the 16 chunks (tiny): x_{c+1} = P_c x_c + S_c.
// One thread per (b,d,n); fully coalesced (n fastest).
// ---------------------------------------------------------------------------
__global__ __launch_bounds__(256)
void chunk_combine(const float* __restrict__ P,
                   const float* __restrict__ S,
                   float* __restrict__ X0)       // (B, NC, D, N) chunk-initial
{
  const int idx = blockIdx.x * 256 + threadIdx.x;   // over B*D*N = 49152
  const int n = idx & (NSTATE - 1);
  const int d = (idx >> 4) % DMODEL;
  const int b = idx / (DMODEL * NSTATE);
  float x = 0.0f;
  for (int c = 0; c < NCHUNK; ++c) {
    const size_t o = (((size_t)b * NCHUNK + c) * DMODEL + d) * NSTATE + n;
    X0[o] = x;
    x = __fmaf_rn(P[o], x, S[o]);
  }
}

// ---------------------------------------------------------------------------
// K4: replay each chunk from its correct initial state, emit y = x . C.
// ---------------------------------------------------------------------------
__global__ __launch_bounds__(256)
void scan_pass2(const float* __restrict__ u,
                const float* __restrict__ A,
                const float* __restrict__ delta,
                const float* __restrict__ Bp,
                const float* __restrict__ Cp,
                const float* __restrict__ X0,
                float* __restrict__ out)          // (B, L, D)
{
  __shared__ float sDelta[CHUNK];
  __shared__ float sB[CHUNK * NSTATE];
  __shared__ float sC[CHUNK * NSTATE];

  const int tid = threadIdx.x;
  const int dt  = blockIdx.x;
  const int c   = blockIdx.y;
  const int b   = blockIdx.z;
  const int d   = dt * DTILE + tid;
  const int l0  = c * CHUNK;
  const size_t rbase = (size_t)b * SEQ + l0;

  if (tid < CHUNK)
    async_ld_b32(lds_off(&sDelta[0]) + tid * 4u, delta + rbase + tid);
  {
    const float* gB = Bp + rbase * NSTATE;
    const float* gC = Cp + rbase * NSTATE;
    for (int i = tid; i < (CHUNK * NSTATE) / 4; i += DTILE) {
      async_ld_b128(lds_off(&sB[0]) + i * 16u, gB + i * 4);
      async_ld_b128(lds_off(&sC[0]) + i * 16u, gC + i * 4);
    }
  }
  wait_async0();
  __syncthreads();

  float Arow[NSTATE];
  {
    const float* Ad = A + (size_t)d * NSTATE;
#pragma unroll
    for (int n = 0; n < NSTATE; ++n) Arow[n] = Ad[n];
  }

  float x[NSTATE];
  {
    const size_t obase = (((size_t)b * NCHUNK + c) * DMODEL + d) * NSTATE;
#pragma unroll
    for (int n = 0; n < NSTATE; ++n) x[n] = X0[obase + n];
  }

  const float* uptr = u + rbase * DMODEL + d;
  float* optr = out + rbase * DMODEL + d;
  for (int t = 0; t < CHUNK; ++t) {
    __builtin_prefetch(uptr + (size_t)(t + 16) * DMODEL, 0, 0);
    const float dl = sDelta[t];
    const float du = dl * uptr[(size_t)t * DMODEL];
    const float* Bt = &sB[t * NSTATE];
    const float* Ct = &sC[t * NSTATE];
    float y = 0.0f;
#pragma unroll
    for (int n = 0; n < NSTATE; ++n) {
      float dA = __expf(dl * Arow[n]);
      x[n] = __fmaf_rn(dA, x[n], du * Bt[n]);
      y    = __fmaf_rn(x[n], Ct[n], y);
    }
    optr[(size_t)t * DMODEL] = y;
  }
}

// ---------------------------------------------------------------------------
// Workspace layout (floats):
//   delta : BL                  =   8192
//   Bp    : BL*N                = 131072
//   Cp    : BL*N                = 131072
//   P     : B*NC*D*N            = 786432
//   S     : B*NC*D*N            = 786432
//   X0    : B*NC*D*N            = 786432
// total ~ 2.63M floats ~ 10.5 MB
// ---------------------------------------------------------------------------
extern "C" void kernel_launch(void* const* d_in, const int* in_sizes, int n_in,
                              void* d_out, int out_size, void* d_ws, size_t ws_size,
                              hipStream_t stream) {
  const float* u       = (const float*)d_in[0];
  const float* A       = (const float*)d_in[1];
  const float* W_B     = (const float*)d_in[2];
  const float* W_C     = (const float*)d_in[3];
  const float* q_delta = (const float*)d_in[4];
  const float* p_delta = (const float*)d_in[5];
  float* out = (float*)d_out;

  float* ws    = (float*)d_ws;
  float* delta = ws;
  float* Bp    = delta + BL;
  float* Cp    = Bp + (size_t)BL * NSTATE;
  float* P     = Cp + (size_t)BL * NSTATE;
  float* S     = P + (size_t)BATCH * NCHUNK * DMODEL * NSTATE;
  float* X0    = S + (size_t)BATCH * NCHUNK * DMODEL * NSTATE;

  // 512 row-tiles, 4 per wave, 4 waves per block -> 32 blocks
  proj_kernel<<<dim3(BL / 16 / (4 * TPW)), dim3(128), 0, stream>>>(
      u, W_B, W_C, q_delta, p_delta, Bp, Cp, delta);

  scan_pass1<<<dim3(DMODEL / DTILE, NCHUNK, BATCH), dim3(DTILE), 0, stream>>>(
      u, A, delta, Bp, P, S);

  chunk_combine<<<dim3((BATCH * DMODEL * NSTATE) / 256), dim3(256), 0, stream>>>(
      P, S, X0);

  scan_pass2<<<dim3(DMODEL / DTILE, NCHUNK, BATCH), dim3(DTILE), 0, stream>>>(
      u, A, delta, Bp, Cp, X0, out);
}